// PseudoTargetModel_48816598287048
// MI455X (gfx1250) — compile-verified
//
#include <hip/hip_runtime.h>
#include <math.h>

#define DIMS   64
#define NTRAIN 4096
#define NQUERY 8192
#define NB     64
#define NOISE  1e-4f
#define LS2    64.0f
#define KSCALE 0.6931471805599453f   /* ln 2 */
#define SPAD   68                    /* SYRK LDS row pad: 16B-aligned float4, conflict-free */

typedef float v2f __attribute__((ext_vector_type(2)));
typedef float v8f __attribute__((ext_vector_type(8)));

// ---------------------------------------------------------------------------
// WMMA helper: D = A(16x4 f32) x B(4x16 f32) + C   (V_WMMA_F32_16X16X4_F32)
// ---------------------------------------------------------------------------
__device__ __forceinline__ v8f wmma4(v2f a, v2f b, v8f c) {
  return __builtin_amdgcn_wmma_f32_16x16x4_f32(
      /*neg_a=*/false, a, /*neg_b=*/false, b,
      /*c_mod=*/(short)0, c, /*reuse_a=*/false, /*reuse_b=*/false);
}

// Fragment loader for NT-GEMM (C = A * B^T, both row-major) from global.
// f32 16x4 A layout: lanes 0-15 -> M=lane, K=k0,k0+1 ; lanes 16-31 -> K=k0+2,k0+3.
__device__ __forceinline__ v2f frag_nt(const float* __restrict__ X, size_t rowStride,
                                       int rowBase, int colBase, int k0, int lane) {
  int kk = k0 + ((lane >> 4) << 1);
  const float* p = X + (size_t)(rowBase + (lane & 15)) * rowStride + colBase + kk;
  v2f v; v.x = p[0]; v.y = p[1];
  return v;
}

// Same fragment pattern out of an LDS-staged panel (row pad SPAD).
__device__ __forceinline__ v2f frag_lds(float (*S)[SPAD], int rowBase, int k0, int lane) {
  int kk = k0 + ((lane >> 4) << 1);
  const float* p = &S[rowBase + (lane & 15)][kk];
  v2f v; v.x = p[0]; v.y = p[1];
  return v;
}

// ---------------------------------------------------------------------------
// Row squared norms
// ---------------------------------------------------------------------------
__global__ void gp_sqnorm(const float* __restrict__ X, float* __restrict__ nrm, int rows) {
  int r = blockIdx.x * blockDim.x + threadIdx.x;
  if (r >= rows) return;
  const float4* p = (const float4*)(X + (size_t)r * DIMS);
  float s = 0.f;
#pragma unroll
  for (int i = 0; i < DIMS / 4; ++i) {
    float4 v = p[i];
    s += v.x * v.x + v.y * v.y + v.z * v.z + v.w * v.w;
  }
  nrm[r] = s;
}

// ---------------------------------------------------------------------------
// K = scale*exp(-0.5*sqdist/ls2) + noise*I   (one wave per 16x16 tile, WMMA)
// ---------------------------------------------------------------------------
__global__ void gp_build_K(const float* __restrict__ tx, const float* __restrict__ nrm,
                           float* __restrict__ K) {
  int bi = blockIdx.x, bj = blockIdx.y;
  int lane = threadIdx.x & 31;
  v8f ce = {}, co = {};                      // two chains -> no WMMA RAW serialization
#pragma unroll
  for (int k0 = 0; k0 < DIMS; k0 += 8) {
    ce = wmma4(frag_nt(tx, DIMS, bi * 16, 0, k0, lane),
               frag_nt(tx, DIMS, bj * 16, 0, k0, lane), ce);
    co = wmma4(frag_nt(tx, DIMS, bi * 16, 0, k0 + 4, lane),
               frag_nt(tx, DIMS, bj * 16, 0, k0 + 4, lane), co);
  }
  v8f c = ce + co;
  int n  = lane & 15;
  int mh = (lane >> 4) * 8;
  int col = bj * 16 + n;
  float ncol = nrm[col];
#pragma unroll
  for (int r = 0; r < 8; ++r) {
    int row = bi * 16 + mh + r;
    float d2 = nrm[row] + ncol - 2.0f * c[r];
    float v  = KSCALE * __expf(-0.5f * d2 * (1.0f / LS2));
    if (row == col) v += NOISE;
    K[(size_t)row * NTRAIN + col] = v;
  }
}

// ---------------------------------------------------------------------------
// Cholesky: diagonal 64x64 block factor (one workgroup of 64 threads)
// ---------------------------------------------------------------------------
__global__ void gp_chol_diag(float* __restrict__ K, int kb) {
  __shared__ float a[NB][NB + 1];
  int t = threadIdx.x;
  int base = kb * NB;
  for (int j = 0; j < NB; ++j) a[t][j] = K[(size_t)(base + t) * NTRAIN + base + j];
  __syncthreads();
  for (int j = 0; j < NB; ++j) {
    if (t == j) a[j][j] = sqrtf(a[j][j]);
    __syncthreads();
    if (t > j) a[t][j] /= a[j][j];
    __syncthreads();
    for (int c2 = j + 1; c2 <= t; ++c2) a[t][c2] -= a[t][j] * a[c2][j];
    __syncthreads();
  }
  for (int j = 0; j <= t; ++j) K[(size_t)(base + t) * NTRAIN + base + j] = a[t][j];
}

// ---------------------------------------------------------------------------
// Cholesky: panel TRSM  L21 = A21 * L11^-T  (64 rows per block, rows in LDS)
// ---------------------------------------------------------------------------
__global__ void gp_chol_trsm(float* __restrict__ K, int kb) {
  __shared__ float Ld[NB][NB + 1];
  __shared__ float rw[NB][NB + 1];
  int t = threadIdx.x;
  int base = kb * NB;
  for (int j = 0; j < NB; ++j) Ld[t][j] = K[(size_t)(base + t) * NTRAIN + base + j];
  int row = base + NB + blockIdx.x * NB + t;
  const float* src = K + (size_t)row * NTRAIN + base;
  for (int j = 0; j < NB; ++j) rw[t][j] = src[j];
  __syncthreads();
  for (int j = 0; j < NB; ++j) {
    float s = rw[t][j];
    for (int p = 0; p < j; ++p) s -= rw[t][p] * Ld[j][p];
    rw[t][j] = s / Ld[j][j];
  }
  float* dst = K + (size_t)row * NTRAIN + base;
  for (int j = 0; j < NB; ++j) dst[j] = rw[t][j];
}

// ---------------------------------------------------------------------------
// Cholesky trailing SYRK: A22 -= L21*L21^T.
// 128 threads = 4 waves per 64x64 output tile; both panels staged in LDS via
// coalesced b128 loads; each wave register-blocks 32x32 (4 independent WMMA
// accumulator chains, 1 LDS fragment load per WMMA).
// ---------------------------------------------------------------------------
__global__ void gp_chol_syrk(float* __restrict__ K, int kb) {
  __shared__ float As[64][SPAD];
  __shared__ float Bs[64][SPAD];
  int bi = blockIdx.x, bj = blockIdx.y;     // 64x64 tiles
  if (bj > bi) return;                      // lower triangle only
  int base = kb * NB;
  int tb   = base + NB;
  int t = threadIdx.x;                      // 128 threads
  int lane = t & 31, wave = t >> 5;
  int rowA = tb + bi * 64, rowB = tb + bj * 64;

  // stage panels: 64 rows x 16 float4 each, fully coalesced
  for (int f = t; f < 64 * 16; f += 128) {
    int r = f >> 4, c4 = (f & 15) << 2;
    *(float4*)&As[r][c4] = *(const float4*)&K[(size_t)(rowA + r) * NTRAIN + base + c4];
    *(float4*)&Bs[r][c4] = *(const float4*)&K[(size_t)(rowB + r) * NTRAIN + base + c4];
  }
  __syncthreads();

  int wr = (wave >> 1) * 32;                // wave's 32x32 sub-tile
  int wc = (wave & 1) * 32;
  v8f c00 = {}, c01 = {}, c10 = {}, c11 = {};
#pragma unroll
  for (int k0 = 0; k0 < NB; k0 += 4) {
    v2f a0 = frag_lds(As, wr,      k0, lane);
    v2f a1 = frag_lds(As, wr + 16, k0, lane);
    v2f b0 = frag_lds(Bs, wc,      k0, lane);
    v2f b1 = frag_lds(Bs, wc + 16, k0, lane);
    c00 = wmma4(a0, b0, c00);
    c01 = wmma4(a0, b1, c01);
    c10 = wmma4(a1, b0, c10);
    c11 = wmma4(a1, b1, c11);
  }

  int n  = lane & 15;
  int mh = (lane >> 4) * 8;
  int r0 = tb + bi * 64 + wr + mh;
  int c0 = tb + bj * 64 + wc + n;
#pragma unroll
  for (int r = 0; r < 8; ++r) {
    K[(size_t)(r0 + r) * NTRAIN + c0]           -= c00[r];
    K[(size_t)(r0 + r) * NTRAIN + c0 + 16]      -= c01[r];
    K[(size_t)(r0 + 16 + r) * NTRAIN + c0]      -= c10[r];
    K[(size_t)(r0 + 16 + r) * NTRAIN + c0 + 16] -= c11[r];
  }
}

// ---------------------------------------------------------------------------
// Forward solve L y = b (in place), single workgroup, left-looking blocked.
// ---------------------------------------------------------------------------
__global__ void gp_fwd_solve(const float* __restrict__ K, float* __restrict__ y) {
  __shared__ float yl[NTRAIN];
  __shared__ float diag[NB][NB + 1];
  __shared__ float pv[NB];
  int t = threadIdx.x;                 // 256 threads
  int wave = t >> 5, lane = t & 31;
  for (int i = t; i < NTRAIN; i += 256) yl[i] = y[i];
  __syncthreads();
  for (int m = 0; m < NTRAIN / NB; ++m) {
    int rb = m * NB;
    for (int idx = t; idx < NB * NB; idx += 256)
      diag[idx >> 6][idx & 63] = K[(size_t)(rb + (idx >> 6)) * NTRAIN + rb + (idx & 63)];
    for (int rr = 0; rr < 8; ++rr) {   // partial dots, coalesced row reads
      int r = wave * 8 + rr;
      float s = 0.f;
      const float* Lr = K + (size_t)(rb + r) * NTRAIN;
      for (int j = lane; j < rb; j += 32) s += Lr[j] * yl[j];
      for (int off = 16; off; off >>= 1) s += __shfl_xor(s, off, 32);
      if (lane == 0) pv[r] = s;
    }
    __syncthreads();
    if (t == 0) {                      // tiny sequential in-block solve
      for (int j = 0; j < NB; ++j) {
        float v = (yl[rb + j] - pv[j]) / diag[j][j];
        yl[rb + j] = v;
        for (int r = j + 1; r < NB; ++r) pv[r] += diag[r][j] * v;
      }
    }
    __syncthreads();
  }
  for (int i = t; i < NTRAIN; i += 256) y[i] = yl[i];
}

// ---------------------------------------------------------------------------
// Backward solve L^T x = y (in place), single workgroup, right-looking blocked.
// ---------------------------------------------------------------------------
__global__ void gp_bwd_solve(const float* __restrict__ K, float* __restrict__ y) {
  __shared__ float yl[NTRAIN];
  __shared__ float diag[NB][NB + 1];
  int t = threadIdx.x;
  for (int i = t; i < NTRAIN; i += 256) yl[i] = y[i];
  __syncthreads();
  for (int m = NTRAIN / NB - 1; m >= 0; --m) {
    int rb = m * NB;
    for (int idx = t; idx < NB * NB; idx += 256)
      diag[idx >> 6][idx & 63] = K[(size_t)(rb + (idx >> 6)) * NTRAIN + rb + (idx & 63)];
    __syncthreads();
    if (t == 0) {
      for (int j = NB - 1; j >= 0; --j) {
        float s = yl[rb + j];
        for (int r = j + 1; r < NB; ++r) s -= diag[r][j] * yl[rb + r];
        yl[rb + j] = s / diag[j][j];
      }
    }
    __syncthreads();
    for (int i = t; i < rb; i += 256) {    // coalesced on i
      float acc = 0.f;
      for (int j = 0; j < NB; ++j) acc += K[(size_t)(rb + j) * NTRAIN + i] * yl[rb + j];
      yl[i] -= acc;
    }
    __syncthreads();
  }
  for (int i = t; i < NTRAIN; i += 256) y[i] = yl[i];
}

// ---------------------------------------------------------------------------
// Fused: Ks tile (WMMA) -> W = Ks*alpha -> rowsum + W @ train_x (WMMA)
//        -> out = x - ((W@tx) - rowsum*x)/ls2
// One wave per 16-row query tile; query A-fragments preloaded (loop-invariant).
// ---------------------------------------------------------------------------
__global__ void gp_fused_grad(const float* __restrict__ x, const float* __restrict__ tx,
                              const float* __restrict__ alpha,
                              const float* __restrict__ nx, const float* __restrict__ nt,
                              float* __restrict__ out) {
  __shared__ float wt[8][16][18];      // per-wave W tile (C-layout -> A-layout bounce)
  int wave = threadIdx.x >> 5, lane = threadIdx.x & 31;
  int rowBase = (blockIdx.x * 8 + wave) * 16;
  int n  = lane & 15;
  int mh = (lane >> 4) * 8;

  float nxr[8];
#pragma unroll
  for (int r = 0; r < 8; ++r) nxr[r] = nx[rowBase + mh + r];

  v2f af[16];                          // loop-invariant query fragments
#pragma unroll
  for (int k0 = 0; k0 < DIMS; k0 += 4)
    af[k0 >> 2] = frag_nt(x, DIMS, rowBase, 0, k0, lane);

  v8f g[4]; g[0] = {}; g[1] = {}; g[2] = {}; g[3] = {};
  float rs[8] = {0, 0, 0, 0, 0, 0, 0, 0};

  for (int c0 = 0; c0 < NTRAIN; c0 += 16) {
    if (c0 + 16 < NTRAIN)              // gfx1250 global_prefetch_b8 of next chunk
      __builtin_prefetch(tx + (size_t)(c0 + 16) * DIMS + lane * 32, 0, 0);
    // ---- distance GEMM: G = x_tile * tx_chunk^T (K = 64), dual chains ----
    v8f ce = {}, co = {};
#pragma unroll
    for (int k0 = 0; k0 < DIMS; k0 += 8) {
      ce = wmma4(af[(k0 >> 2)],     frag_nt(tx, DIMS, c0, 0, k0,     lane), ce);
      co = wmma4(af[(k0 >> 2) + 1], frag_nt(tx, DIMS, c0, 0, k0 + 4, lane), co);
    }
    v8f c = ce + co;
    // ---- transform to W = scale*exp(-0.5*d2/ls2)*alpha, stash in LDS ----
    float ncol = nt[c0 + n];
    float av   = alpha[c0 + n];
#pragma unroll
    for (int r = 0; r < 8; ++r) {
      float d2 = nxr[r] + ncol - 2.0f * c[r];
      float w  = KSCALE * __expf(-0.5f * d2 * (1.0f / LS2)) * av;
      rs[r] += w;
      wt[wave][mh + r][n] = w;
    }
    __syncthreads();
    // ---- grad GEMM: G2[16x64] += W[16x16] @ tx_chunk[16x64] (K = 16) ----
#pragma unroll
    for (int nc = 0; nc < 4; ++nc) {
#pragma unroll
      for (int k0 = 0; k0 < 16; k0 += 4) {
        int kk = k0 + ((lane >> 4) << 1);
        v2f a, b;
        a.x = wt[wave][n][kk];
        a.y = wt[wave][n][kk + 1];
        b.x = tx[(size_t)(c0 + kk) * DIMS + nc * 16 + n];
        b.y = tx[(size_t)(c0 + kk + 1) * DIMS + nc * 16 + n];
        g[nc] = wmma4(a, b, g[nc]);
      }
    }
    __syncthreads();
  }

  // rowsum reduce across the 16 lanes of each half (xor bits 0..3 only)
#pragma unroll
  for (int r = 0; r < 8; ++r)
    for (int off = 8; off; off >>= 1) rs[r] += __shfl_xor(rs[r], off, 32);

  // out = x - ((W@tx) - rowsum*x) / ls2
#pragma unroll
  for (int nc = 0; nc < 4; ++nc) {
#pragma unroll
    for (int r = 0; r < 8; ++r) {
      int row = rowBase + mh + r;
      int d   = nc * 16 + n;
      float xv = x[(size_t)row * DIMS + d];
      out[(size_t)row * DIMS + d] = xv - (g[nc][r] - rs[r] * xv) * (1.0f / LS2);
    }
  }
}

// ---------------------------------------------------------------------------
extern "C" void kernel_launch(void* const* d_in, const int* in_sizes, int n_in,
                              void* d_out, int out_size, void* d_ws, size_t ws_size,
                              hipStream_t stream) {
  (void)in_sizes; (void)n_in; (void)out_size; (void)ws_size;
  const float* x  = (const float*)d_in[0];
  const float* tx = (const float*)d_in[1];
  const float* ty = (const float*)d_in[2];
  float* out = (float*)d_out;

  float* Kd = (float*)d_ws;                       // 4096*4096 f32 = 64 MB
  float* nt = Kd + (size_t)NTRAIN * NTRAIN;       // train norms
  float* nx = nt + NTRAIN;                        // query norms
  float* al = nx + NQUERY;                        // alpha (solved in place)

  gp_sqnorm<<<(NTRAIN + 255) / 256, 256, 0, stream>>>(tx, nt, NTRAIN);
  gp_sqnorm<<<(NQUERY + 255) / 256, 256, 0, stream>>>(x, nx, NQUERY);

  gp_build_K<<<dim3(NTRAIN / 16, NTRAIN / 16), 32, 0, stream>>>(tx, nt, Kd);

  for (int kb = 0; kb < NTRAIN / NB; ++kb) {
    gp_chol_diag<<<1, NB, 0, stream>>>(Kd, kb);
    int below = NTRAIN - NB * (kb + 1);
    if (below > 0) {
      gp_chol_trsm<<<below / NB, NB, 0, stream>>>(Kd, kb);
      int tt = below / 64;
      gp_chol_syrk<<<dim3(tt, tt), 128, 0, stream>>>(Kd, kb);
    }
  }

  hipMemcpyAsync(al, ty, NTRAIN * sizeof(float), hipMemcpyDeviceToDevice, stream);
  gp_fwd_solve<<<1, 256, 0, stream>>>(Kd, al);
  gp_bwd_solve<<<1, 256, 0, stream>>>(Kd, al);

  gp_fused_grad<<<NQUERY / 128, 256, 0, stream>>>(x, tx, al, nx, nt, out);
}